// FBCritic_50319836840675
// MI455X (gfx1250) — compile-verified
//
#include <hip/hip_runtime.h>

typedef __attribute__((ext_vector_type(2))) float v2f;
typedef __attribute__((ext_vector_type(8))) float v8f;

#define NUM_OBS 100000
#define NUM_ACT 10
#define RD 64          // REPR_DIM (K)
#define TILE 128       // 128x128 output tile per block
#define KH 32          // K processed per LDS stage
#define LDS_PITCH 36   // KH + 4 pad -> conflict-free float2 reads (stride mod 64 steps banks by 4)

__global__ __launch_bounds__(256)
void fb_outer_wmma_f32(const int* __restrict__ obs, const int* __restrict__ act,
                       const int* __restrict__ fobs, const int* __restrict__ fact,
                       const float* __restrict__ Wf, const float* __restrict__ Wb,
                       float* __restrict__ out, int Bn)
{
    __shared__ float As[TILE * LDS_PITCH];   // forward panel  (this block's M-tile rows)
    __shared__ float Bs[TILE * LDS_PITCH];   // backward panel (this block's N-tile rows)
    __shared__ int   rowF[TILE];
    __shared__ int   rowB[TILE];

    const int t  = threadIdx.x;
    const int w  = t >> 5;        // wave id 0..7 (wave32)
    const int l  = t & 31;        // lane
    const int m0 = blockIdx.y * TILE;
    const int n0 = blockIdx.x * TILE;

    // ravel_multi_index(..., mode='clip') for this block's 128 A-rows / 128 B-rows
    if (t < TILE) {
        int o = obs[m0 + t];  o = min(max(o, 0), NUM_OBS - 1);
        int a = act[m0 + t];  a = min(max(a, 0), NUM_ACT - 1);
        rowF[t] = o * NUM_ACT + a;
    } else {
        int r = t - TILE;
        int o = fobs[n0 + r]; o = min(max(o, 0), NUM_OBS - 1);
        int a = fact[n0 + r]; a = min(max(a, 0), NUM_ACT - 1);
        rowB[r] = o * NUM_ACT + a;
    }
    __syncthreads();

    // 8 waves tile the 128x128 block as 4 (M) x 2 (N); each wave owns 32x64:
    // 2 (M) x 4 (N) subtiles of 16x16 -> 6 LDS fragment loads per 8 WMMAs.
    const int wm = w & 3;            // M tile index  -> rows  wm*32 .. +31
    const int wn = w >> 2;           // N tile index  -> cols  wn*64 .. +63

    v8f acc[2][4];
    #pragma unroll
    for (int i = 0; i < 2; ++i)
        #pragma unroll
        for (int j = 0; j < 4; ++j)
            #pragma unroll
            for (int e = 0; e < 8; ++e) acc[i][j][e] = 0.0f;

    const int r16 = l & 15;          // M (A frag) / N (B frag) within the 16x16 tile
    const int kb  = (l >> 4) * 2;    // lanes 16-31 hold K=2,3 of each 4-wide K step
    const int aBase = (wm * 32 + r16) * LDS_PITCH + kb;
    const int bBase = (wn * 64 + r16) * LDS_PITCH + kb;

    for (int h = 0; h < 2; ++h) {    // two K halves of 32
        // stage 128x32 of each gathered panel into LDS: 4 float4 per thread per panel
        #pragma unroll
        for (int i = 0; i < 4; ++i) {
            int idx = t + i * 256;           // (row, c4) over 128 x 8 float4s
            int row = idx >> 3;
            int c4  = idx & 7;
            float4 a4 = *(const float4*)(Wf + (size_t)rowF[row] * RD + h * KH + c4 * 4);
            *(float4*)(&As[row * LDS_PITCH + c4 * 4]) = a4;
            float4 b4 = *(const float4*)(Wb + (size_t)rowB[row] * RD + h * KH + c4 * 4);
            *(float4*)(&Bs[row * LDS_PITCH + c4 * 4]) = b4;
        }
        __syncthreads();

        #pragma unroll
        for (int kk = 0; kk < KH; kk += 4) {
            // 6 fragment loads feed 8 independent WMMAs
            v2f a0 = *(const v2f*)(&As[aBase + kk]);
            v2f a1 = *(const v2f*)(&As[aBase + 16 * LDS_PITCH + kk]);
            v2f b0 = *(const v2f*)(&Bs[bBase + kk]);
            v2f b1 = *(const v2f*)(&Bs[bBase + 16 * LDS_PITCH + kk]);
            v2f b2 = *(const v2f*)(&Bs[bBase + 32 * LDS_PITCH + kk]);
            v2f b3 = *(const v2f*)(&Bs[bBase + 48 * LDS_PITCH + kk]);

            acc[0][0] = __builtin_amdgcn_wmma_f32_16x16x4_f32(false, a0, false, b0, (short)0, acc[0][0], false, false);
            acc[0][1] = __builtin_amdgcn_wmma_f32_16x16x4_f32(false, a0, false, b1, (short)0, acc[0][1], false, false);
            acc[0][2] = __builtin_amdgcn_wmma_f32_16x16x4_f32(false, a0, false, b2, (short)0, acc[0][2], false, false);
            acc[0][3] = __builtin_amdgcn_wmma_f32_16x16x4_f32(false, a0, false, b3, (short)0, acc[0][3], false, false);
            acc[1][0] = __builtin_amdgcn_wmma_f32_16x16x4_f32(false, a1, false, b0, (short)0, acc[1][0], false, false);
            acc[1][1] = __builtin_amdgcn_wmma_f32_16x16x4_f32(false, a1, false, b1, (short)0, acc[1][1], false, false);
            acc[1][2] = __builtin_amdgcn_wmma_f32_16x16x4_f32(false, a1, false, b2, (short)0, acc[1][2], false, false);
            acc[1][3] = __builtin_amdgcn_wmma_f32_16x16x4_f32(false, a1, false, b3, (short)0, acc[1][3], false, false);
        }
        __syncthreads();
    }

    // D layout: VGPR e -> M = base + e + 8*(lane>=16), N = base + (lane&15).
    // Non-temporal: 256 MB streamed out once, keep L2 for the hot W panels.
    #pragma unroll
    for (int i = 0; i < 2; ++i) {
        const int mrow = m0 + wm * 32 + i * 16 + ((l >> 4) ? 8 : 0);
        #pragma unroll
        for (int j = 0; j < 4; ++j) {
            float* o = out + (size_t)mrow * Bn + (n0 + wn * 64 + j * 16 + r16);
            #pragma unroll
            for (int e = 0; e < 8; ++e) {
                __builtin_nontemporal_store(acc[i][j][e], o + (size_t)e * Bn);
            }
        }
    }
}

extern "C" void kernel_launch(void* const* d_in, const int* in_sizes, int n_in,
                              void* d_out, int out_size, void* d_ws, size_t ws_size,
                              hipStream_t stream) {
    const int*   obs  = (const int*)d_in[0];
    const int*   act  = (const int*)d_in[1];
    const int*   fobs = (const int*)d_in[2];
    const int*   fact = (const int*)d_in[3];
    const float* Wf   = (const float*)d_in[4];
    const float* Wb   = (const float*)d_in[5];
    float*       out  = (float*)d_out;

    const int Bn = in_sizes[0];          // 8192
    dim3 grid(Bn / TILE, Bn / TILE);     // 64 x 64 blocks of 128x128
    fb_outer_wmma_f32<<<grid, dim3(256), 0, stream>>>(obs, act, fobs, fact, Wf, Wb, out, Bn);
}